// Discriminator_5763846111635
// MI455X (gfx1250) — compile-verified
//
#include <hip/hip_runtime.h>

typedef float v2f __attribute__((ext_vector_type(2)));
typedef float v4f __attribute__((ext_vector_type(4)));
typedef float v8f __attribute__((ext_vector_type(8)));

static constexpr int Bn = 512, Tn = 4096, Dn = 8, Hn = 16;

static __device__ __forceinline__ v8f wmma4(v2f a, v2f b, v8f c) {
  // D = A(16x4,f32) * B(4x16,f32) + C(16x16,f32)
  return __builtin_amdgcn_wmma_f32_16x16x4_f32(false, a, false, b, (short)0, c,
                                               false, false);
}

static __device__ __forceinline__ float fast_rcp(float x) {
#if __has_builtin(__builtin_amdgcn_rcpf)
  return __builtin_amdgcn_rcpf(x);  // v_rcp_f32
#else
  return 1.0f / x;
#endif
}

static __device__ __forceinline__ float fast_tanh(float x) {
#if __has_builtin(__builtin_amdgcn_tanhf)
  return __builtin_amdgcn_tanhf(x);  // v_tanh_f32 (new on gfx1250)
#else
  // tanh(x) = 1 - 2/(exp(2x)+1); exp->inf / ->0 limits give +-1 correctly
  float e = __expf(2.0f * x);
  return 1.0f - 2.0f * fast_rcp(e + 1.0f);
#endif
}

static __device__ __forceinline__ float lipswish(float x) {
  // 0.909 * x * sigmoid(x), sigmoid via v_exp + v_rcp (no IEEE div expansion)
  return 0.909f * x * fast_rcp(1.0f + __expf(-x));
}

__global__ __launch_bounds__(32) void ncde_scan_kernel(
    const float* __restrict__ ts, const float* __restrict__ ys,
    const float* __restrict__ iW1, const float* __restrict__ ib1,
    const float* __restrict__ iW2, const float* __restrict__ ib2,
    const float* __restrict__ vW1, const float* __restrict__ vb1,
    const float* __restrict__ vW2, const float* __restrict__ vb2,
    const float* __restrict__ cW1, const float* __restrict__ cb1,
    const float* __restrict__ cW2, const float* __restrict__ cb2,
    const float* __restrict__ rW, const float* __restrict__ rb,
    float* __restrict__ out) {
  __shared__ float s_y[16 * 16];    // C->A staging (reused for y, h1, h2)
  __shared__ float s_g[16 * 128];   // diffusion output tile
  __shared__ float s_dx[16 * 8];    // control increment
  __shared__ float s_add[16 * 16];  // einsum result / init scratch

  const int lane = threadIdx.x;  // 0..31
  const int m0 = blockIdx.x * 16;
  const int col = lane & 15;
  const int half = lane >> 4;

  // ---------------- preload weights in WMMA B-layout ----------------
  // B chunk k, component v, lane L holds B[K = 4k + 2*(L/16) + v][N = L%16].
  v2f Bv1[4], Bv2[4], Bc1[4], Bc2[8][4];
#pragma unroll
  for (int k = 0; k < 4; ++k) {
    const int kk = 4 * k + 2 * half;
    Bv1[k] = v2f{vW1[col * 17 + 1 + kk], vW1[col * 17 + 2 + kk]};  // skip t col
    Bc1[k] = v2f{cW1[col * 17 + 1 + kk], cW1[col * 17 + 2 + kk]};
    Bv2[k] = v2f{vW2[col * 16 + kk], vW2[col * 16 + kk + 1]};
#pragma unroll
    for (int o = 0; o < 8; ++o)
      Bc2[o][k] = v2f{cW2[(o * 16 + col) * 16 + kk],
                      cW2[(o * 16 + col) * 16 + kk + 1]};
  }
  const float vb1c = vb1[col], vb2c = vb2[col], cb1c = cb1[col];
  const float vw1t = vW1[col * 17], cw1t = cW1[col * 17];
  float cb2c[8];
#pragma unroll
  for (int o = 0; o < 8; ++o) cb2c[o] = cb2[o * 16 + col];
  const float t0 = ts[0];

  // ---------------- initial hidden state y0 ----------------
  // y0 = relu([t0, ys[:,0]] @ iW1.T + ib1) @ iW2.T + ib2
  {
    const int m = col, hb = 8 * half;
    float yrow[8];
#pragma unroll
    for (int j = 0; j < 8; ++j)
      yrow[j] = ys[(size_t)(m0 + m) * Tn * Dn + j];
#pragma unroll
    for (int w = 0; w < 8; ++w) {
      const int ww = hb + w;
      float a = ib1[ww] + t0 * iW1[ww * 9];
#pragma unroll
      for (int j = 0; j < 8; ++j) a += yrow[j] * iW1[ww * 9 + 1 + j];
      s_add[m * 16 + ww] = fmaxf(a, 0.0f);
    }
    __syncthreads();
#pragma unroll
    for (int h = 0; h < 8; ++h) {
      const int hh = hb + h;
      float a = ib2[hh];
#pragma unroll
      for (int w = 0; w < 16; ++w) a += s_add[m * 16 + w] * iW2[hh * 16 + w];
      s_y[m * 16 + hh] = a;
    }
    __syncthreads();
  }

  // y accumulator in WMMA C/D layout: reg r, lane holds [m0 + r + 8*half][col]
  v8f yc;
#pragma unroll
  for (int r = 0; r < 8; ++r) yc[r] = s_y[(r + 8 * half) * 16 + col];

  // readout at t0 (SaveAt t0)
  if (lane < 16) {
    float a = rb[0];
#pragma unroll
    for (int h = 0; h < 16; ++h) a += s_y[lane * 16 + h] * rW[h];
    out[(size_t)(m0 + lane) * 2 + 0] = a;
  }

  // per-lane streaming pointer for ys rows: lane covers (m = lane/2, 4 of 8 d)
  const int ml = lane >> 1, dq = (lane & 1) * 4;
  const float* ysrow = ys + (size_t)(m0 + ml) * Tn * Dn + dq;
  v4f yprev = *(const v4f*)(ysrow);  // ys[:, 0]

  // ---------------- Euler scan over T-1 steps ----------------
  for (int n = 0; n < Tn - 1; ++n) {
    const float tcur = t0 + (float)n;

    // control increment dX = ys[n+1] - ys[n], staged to LDS
    v4f ynext = *(const v4f*)(ysrow + (size_t)(n + 1) * Dn);
    const int pn = (n + 2 < Tn) ? (n + 2) : (Tn - 1);
    __builtin_prefetch(ysrow + (size_t)pn * Dn, 0, 1);
    v4f dx = ynext - yprev;
    yprev = ynext;
    *(v4f*)&s_dx[ml * 8 + dq] = dx;

    // ---- y: C/D layout -> A layout via LDS ----
#pragma unroll
    for (int r = 0; r < 8; ++r) s_y[(r + 8 * half) * 16 + col] = yc[r];
    __syncthreads();
    v2f Ay[4];
#pragma unroll
    for (int k = 0; k < 4; ++k)
      Ay[k] = *(const v2f*)&s_y[col * 16 + 4 * k + 2 * half];

    // ---- drift hidden: h1 = lipswish(z @ vW1.T + vb1) ----
    v8f h1;
    {
      const float bias = vb1c + tcur * vw1t;
#pragma unroll
      for (int r = 0; r < 8; ++r) h1[r] = bias;
#pragma unroll
      for (int k = 0; k < 4; ++k) h1 = wmma4(Ay[k], Bv1[k], h1);
#pragma unroll
      for (int r = 0; r < 8; ++r) h1[r] = lipswish(h1[r]);
    }
    __syncthreads();
#pragma unroll
    for (int r = 0; r < 8; ++r) s_y[(r + 8 * half) * 16 + col] = h1[r];
    __syncthreads();
    v2f Ah[4];
#pragma unroll
    for (int k = 0; k < 4; ++k)
      Ah[k] = *(const v2f*)&s_y[col * 16 + 4 * k + 2 * half];

    // ---- drift f = tanh(h1 @ vW2.T + vb2) ----
    v8f f;
#pragma unroll
    for (int r = 0; r < 8; ++r) f[r] = vb2c;
#pragma unroll
    for (int k = 0; k < 4; ++k) f = wmma4(Ah[k], Bv2[k], f);
#pragma unroll
    for (int r = 0; r < 8; ++r) f[r] = fast_tanh(f[r]);

    // ---- diffusion hidden: h2 = lipswish(z @ cW1.T + cb1) ----
    v8f h2;
    {
      const float bias = cb1c + tcur * cw1t;
#pragma unroll
      for (int r = 0; r < 8; ++r) h2[r] = bias;
#pragma unroll
      for (int k = 0; k < 4; ++k) h2 = wmma4(Ay[k], Bc1[k], h2);
#pragma unroll
      for (int r = 0; r < 8; ++r) h2[r] = lipswish(h2[r]);
    }
    __syncthreads();
#pragma unroll
    for (int r = 0; r < 8; ++r) s_y[(r + 8 * half) * 16 + col] = h2[r];
    __syncthreads();
    v2f Ag[4];
#pragma unroll
    for (int k = 0; k < 4; ++k)
      Ag[k] = *(const v2f*)&s_y[col * 16 + 4 * k + 2 * half];

    // ---- diffusion g = tanh(h2 @ cW2.T + cb2): 8 output tiles of 16x16 ----
#pragma unroll
    for (int o = 0; o < 8; ++o) {
      v8f g;
#pragma unroll
      for (int r = 0; r < 8; ++r) g[r] = cb2c[o];
#pragma unroll
      for (int k = 0; k < 4; ++k) g = wmma4(Ag[k], Bc2[o][k], g);
#pragma unroll
      for (int r = 0; r < 8; ++r)
        s_g[(r + 8 * half) * 128 + o * 16 + col] = fast_tanh(g[r]);
    }
    __syncthreads();

    // ---- einsum('bhd,bd->bh'): lane handles m=col, h in [8*half, 8*half+8) --
    {
      const int me = col, hb = 8 * half;
      v4f dx0 = *(const v4f*)&s_dx[me * 8];
      v4f dx1 = *(const v4f*)&s_dx[me * 8 + 4];
      float acc[8];
#pragma unroll
      for (int h = 0; h < 8; ++h) {
        v4f ga = *(const v4f*)&s_g[me * 128 + (hb + h) * 8];
        v4f gb = *(const v4f*)&s_g[me * 128 + (hb + h) * 8 + 4];
        acc[h] = ga.x * dx0.x + ga.y * dx0.y + ga.z * dx0.z + ga.w * dx0.w +
                 gb.x * dx1.x + gb.y * dx1.y + gb.z * dx1.z + gb.w * dx1.w;
      }
#pragma unroll
      for (int h = 0; h < 8; ++h) s_add[me * 16 + hb + h] = acc[h];
    }
    __syncthreads();

    // ---- y += f + einsum, back in C/D layout ----
#pragma unroll
    for (int r = 0; r < 8; ++r)
      yc[r] += f[r] + s_add[(r + 8 * half) * 16 + col];
    __syncthreads();
  }

  // ---------------- readout at t1 ----------------
#pragma unroll
  for (int r = 0; r < 8; ++r) s_y[(r + 8 * half) * 16 + col] = yc[r];
  __syncthreads();
  if (lane < 16) {
    float a = rb[0];
#pragma unroll
    for (int h = 0; h < 16; ++h) a += s_y[lane * 16 + h] * rW[h];
    out[(size_t)(m0 + lane) * 2 + 1] = a;
  }
}

extern "C" void kernel_launch(void* const* d_in, const int* in_sizes, int n_in,
                              void* d_out, int out_size, void* d_ws,
                              size_t ws_size, hipStream_t stream) {
  (void)in_sizes; (void)n_in; (void)out_size; (void)d_ws; (void)ws_size;
  const float* ts  = (const float*)d_in[0];
  const float* ys  = (const float*)d_in[1];
  const float* iW1 = (const float*)d_in[2];
  const float* ib1 = (const float*)d_in[3];
  const float* iW2 = (const float*)d_in[4];
  const float* ib2 = (const float*)d_in[5];
  const float* vW1 = (const float*)d_in[6];
  const float* vb1 = (const float*)d_in[7];
  const float* vW2 = (const float*)d_in[8];
  const float* vb2 = (const float*)d_in[9];
  const float* cW1 = (const float*)d_in[10];
  const float* cb1 = (const float*)d_in[11];
  const float* cW2 = (const float*)d_in[12];
  const float* cb2 = (const float*)d_in[13];
  const float* rW  = (const float*)d_in[14];
  const float* rb  = (const float*)d_in[15];
  float* out = (float*)d_out;

  // 32 batch tiles of 16 rows, one wave32 per tile.
  ncde_scan_kernel<<<dim3(Bn / 16), dim3(32), 0, stream>>>(
      ts, ys, iW1, ib1, iW2, ib2, vW1, vb1, vW2, vb2, cW1, cb1, cW2, cb2, rW,
      rb, out);
}